// DeformableBiomarkerAttention_58188216926550
// MI455X (gfx1250) — compile-verified
//
#include <hip/hip_runtime.h>
#include <hip/hip_bf16.h>

#define E_DIM   768
#define H_HEADS 12
#define HD      64
#define BATCH   64
#define FULLN   513
#define NB      32
#define GRID    8

typedef __attribute__((ext_vector_type(2))) float v2f;
typedef __attribute__((ext_vector_type(4))) float v4f;
typedef __attribute__((ext_vector_type(8))) float v8f;

static __device__ __forceinline__ int clampi(int v, int lo, int hi) {
    return v < lo ? lo : (v > hi ? hi : v);
}

// ---------------------------------------------------------------------------
// Kernel 1: trilinear border sample. One block per (b,p) point; 256 threads
// stride over the 768 channels. Gathers 8 corner token rows from x.
// ---------------------------------------------------------------------------
__global__ void sample_kernel(const float* __restrict__ x,
                              const float* __restrict__ base,
                              const float* __restrict__ offs,
                              float* __restrict__ sampled) {
    const int bp = blockIdx.x;            // 0..2047
    const int b = bp >> 5;
    const int p = bp & 31;

    float cx = base[p * 3 + 0] + offs[bp * 3 + 0];
    float cy = base[p * 3 + 1] + offs[bp * 3 + 1];
    float cz = base[p * 3 + 2] + offs[bp * 3 + 2];
    cx = fminf(fmaxf(cx, -1.f), 1.f);
    cy = fminf(fmaxf(cy, -1.f), 1.f);
    cz = fminf(fmaxf(cz, -1.f), 1.f);

    const float ix = (cx + 1.f) * 0.5f * (GRID - 1);
    const float iy = (cy + 1.f) * 0.5f * (GRID - 1);
    const float iz = (cz + 1.f) * 0.5f * (GRID - 1);
    const float fx = floorf(ix), fy = floorf(iy), fz = floorf(iz);
    const float wx = ix - fx, wy = iy - fy, wz = iz - fz;
    const int x0 = clampi((int)fx, 0, GRID - 1), x1 = clampi((int)fx + 1, 0, GRID - 1);
    const int y0 = clampi((int)fy, 0, GRID - 1), y1 = clampi((int)fy + 1, 0, GRID - 1);
    const int z0 = clampi((int)fz, 0, GRID - 1), z1 = clampi((int)fz + 1, 0, GRID - 1);

    const float w000 = (1.f - wz) * (1.f - wy) * (1.f - wx);
    const float w001 = (1.f - wz) * (1.f - wy) * wx;
    const float w010 = (1.f - wz) * wy * (1.f - wx);
    const float w011 = (1.f - wz) * wy * wx;
    const float w100 = wz * (1.f - wy) * (1.f - wx);
    const float w101 = wz * (1.f - wy) * wx;
    const float w110 = wz * wy * (1.f - wx);
    const float w111 = wz * wy * wx;

    // token row pointers: feat_map[b,c,z,y,x] = x[b, 1 + (z*8+y)*8+x, c]
    const size_t rowB = (size_t)b * FULLN + 1;
    const float* p000 = x + (rowB + ((z0 * GRID + y0) * GRID + x0)) * E_DIM;
    const float* p001 = x + (rowB + ((z0 * GRID + y0) * GRID + x1)) * E_DIM;
    const float* p010 = x + (rowB + ((z0 * GRID + y1) * GRID + x0)) * E_DIM;
    const float* p011 = x + (rowB + ((z0 * GRID + y1) * GRID + x1)) * E_DIM;
    const float* p100 = x + (rowB + ((z1 * GRID + y0) * GRID + x0)) * E_DIM;
    const float* p101 = x + (rowB + ((z1 * GRID + y0) * GRID + x1)) * E_DIM;
    const float* p110 = x + (rowB + ((z1 * GRID + y1) * GRID + x0)) * E_DIM;
    const float* p111 = x + (rowB + ((z1 * GRID + y1) * GRID + x1)) * E_DIM;

    float* dst = sampled + (size_t)bp * E_DIM;
    for (int c = threadIdx.x; c < E_DIM; c += blockDim.x) {
        float acc = w000 * p000[c] + w001 * p001[c] + w010 * p010[c] + w011 * p011[c]
                  + w100 * p100[c] + w101 * p101[c] + w110 * p110[c] + w111 * p111[c];
        dst[c] = acc;
    }
}

// ---------------------------------------------------------------------------
// Kernel 2: fp32 WMMA GEMM:  C[M,N] = A[M,K] @ W[N,K]^T + bias[N]
// One wave (32 threads) per 64x32 C tile: 4x2 grid of 16x16 WMMA tiles,
// K-loop in steps of 4 using V_WMMA_F32_16X16X4_F32.
// 8 v8f accumulators = 64 VGPRs -> no spills (prev 4x4 version spilled).
//
// f32 A-frag layout (ISA 7.12.2): lane l<16 holds {A[m][k0],A[m][k0+1]} with
// m=l, k0=0; lanes 16..31 same m with k0=2.  B(4x16)=W^T consumed per-lane as
// {W[n][k0], W[n][k0+1]}, n = lane&15 — a contiguous 8B load from row-major W.
// ---------------------------------------------------------------------------
__global__ __launch_bounds__(32) void gemm_wmma_f32(
        const float* __restrict__ A,
        const float* __restrict__ W,
        const float* __restrict__ bias,
        float* __restrict__ C,
        int M, int N, int K) {
    const int lane  = threadIdx.x;
    const int l15   = lane & 15;
    const int kh    = lane >> 4;           // 0 or 1 -> k offset 0 or 2
    const int mBase = blockIdx.y * 64;
    const int nBase = blockIdx.x * 32;

    // fixed per-lane base pointers; K offsets folded into load immediates
    const float* aP[4];
    const float* bP[2];
#pragma unroll
    for (int i = 0; i < 4; ++i)
        aP[i] = A + (size_t)(mBase + 16 * i + l15) * K + 2 * kh;
#pragma unroll
    for (int j = 0; j < 2; ++j)
        bP[j] = W + (size_t)(nBase + 16 * j + l15) * K + 2 * kh;

    v8f acc[4][2] = {};

#pragma unroll 4
    for (int k0 = 0; k0 < K; k0 += 4) {
        v2f a[4], b[2];
#pragma unroll
        for (int i = 0; i < 4; ++i) a[i] = *(const v2f*)(aP[i] + k0);
#pragma unroll
        for (int j = 0; j < 2; ++j) b[j] = *(const v2f*)(bP[j] + k0);
#pragma unroll
        for (int i = 0; i < 4; ++i)
#pragma unroll
            for (int j = 0; j < 2; ++j)
                acc[i][j] = __builtin_amdgcn_wmma_f32_16x16x4_f32(
                    /*neg_a=*/false, a[i], /*neg_b=*/false, b[j],
                    /*c_mod=*/(short)0, acc[i][j],
                    /*reuse_a=*/false, /*reuse_b=*/false);
    }

    // C/D layout: VGPR r -> row (r + 8*kh), col = lane&15 within the 16x16 tile
#pragma unroll
    for (int j = 0; j < 2; ++j) {
        const int col = nBase + 16 * j + l15;
        const float bv = bias[col];
#pragma unroll
        for (int i = 0; i < 4; ++i) {
            const int rowBase = mBase + 16 * i + 8 * kh;
#pragma unroll
            for (int r = 0; r < 8; ++r) {
                C[(size_t)(rowBase + r) * N + col] = acc[i][j][r] + bv;
            }
        }
    }
}

// ---------------------------------------------------------------------------
// Kernel 3: attention.  One wave per (b, h).  lane = key index p (0..31).
// scores -> softmax via wave shfl reductions -> ctx (each lane owns 2 dims).
// kv holds [k | v] rows of length 1536 per (b,p).
// ---------------------------------------------------------------------------
__global__ __launch_bounds__(32) void attn_kernel(
        const float* __restrict__ q,    // (64, 768)
        const float* __restrict__ kv,   // (2048, 1536)
        float* __restrict__ ctx) {      // (64, 768)
    const int bh = blockIdx.x;
    const int b = bh / H_HEADS;
    const int h = bh % H_HEADS;
    const int lane = threadIdx.x;       // = p

    __shared__ float attn_s[NB];

    const float* qv   = q + (size_t)b * E_DIM + h * HD;
    const float* krow = kv + (size_t)(b * NB + lane) * (2 * E_DIM) + h * HD;

    float s = 0.f;
#pragma unroll
    for (int d = 0; d < HD; ++d) s += qv[d] * krow[d];
    s *= 0.125f;                         // 1/sqrt(64)

    float mx = s;
#pragma unroll
    for (int off = 16; off > 0; off >>= 1) mx = fmaxf(mx, __shfl_xor(mx, off, 32));
    const float e = __expf(s - mx);
    float sum = e;
#pragma unroll
    for (int off = 16; off > 0; off >>= 1) sum += __shfl_xor(sum, off, 32);
    attn_s[lane] = e / sum;
    __syncthreads();

    const float* vbase = kv + (size_t)(b * NB) * (2 * E_DIM) + E_DIM + h * HD;
    float c0 = 0.f, c1 = 0.f;
#pragma unroll
    for (int p = 0; p < NB; ++p) {
        const float a = attn_s[p];
        const float* vr = vbase + (size_t)p * (2 * E_DIM);
        c0 += a * vr[lane];
        c1 += a * vr[lane + 32];
    }
    float* dst = ctx + (size_t)b * E_DIM + h * HD;
    dst[lane]      = c0;
    dst[lane + 32] = c1;
}

// ---------------------------------------------------------------------------
// Kernel 4: broadcast * confidence — the 100 MB streaming-store kernel
// (HBM roofline term).  out[b,n,c] = attn_out[b,c] * conf[b].
// Nontemporal float4 stores: output is write-once, never re-read.
// ---------------------------------------------------------------------------
__global__ void bcast_kernel(const float* __restrict__ ao,    // (64, 768)
                             const float* __restrict__ conf,  // (64,)
                             float* __restrict__ out) {
    const int b = blockIdx.y;
    const int inner = FULLN * (E_DIM / 4);       // 98496 float4 per batch row
    const int i = blockIdx.x * blockDim.x + threadIdx.x;
    if (i >= inner) return;
    const int c4 = i % (E_DIM / 4);
    const float cf = conf[b];
    v4f a = *(const v4f*)(ao + (size_t)b * E_DIM + (size_t)c4 * 4);
    a *= cf;
    __builtin_nontemporal_store(a, (v4f*)(out + ((size_t)b * inner + i) * 4));
}

// ---------------------------------------------------------------------------
extern "C" void kernel_launch(void* const* d_in, const int* in_sizes, int n_in,
                              void* d_out, int out_size, void* d_ws, size_t ws_size,
                              hipStream_t stream) {
    const float* x          = (const float*)d_in[0];   // (64, 513, 768)
    const float* bio_embed  = (const float*)d_in[1];   // (64, 768)
    const float* base       = (const float*)d_in[2];   // (32, 3)
    const float* offsets    = (const float*)d_in[3];   // (64, 32, 3)
    const float* confidence = (const float*)d_in[4];   // (64, 1)
    const float* sp_w       = (const float*)d_in[5];   // (768, 768)
    const float* sp_b       = (const float*)d_in[6];   // (768,)
    const float* in_w       = (const float*)d_in[7];   // (2304, 768)
    const float* in_b       = (const float*)d_in[8];   // (2304,)
    const float* out_w      = (const float*)d_in[9];   // (768, 768)
    const float* out_b      = (const float*)d_in[10];  // (768,)
    float* out = (float*)d_out;

    // workspace layout (floats)
    float* ws = (float*)d_ws;
    const size_t MP = (size_t)BATCH * NB;              // 2048
    float* sampled  = ws;                              // 2048*768
    float* sproj    = sampled + MP * E_DIM;            // 2048*768
    float* kvbuf    = sproj + MP * E_DIM;              // 2048*1536
    float* qbuf     = kvbuf + MP * 2 * E_DIM;          // 64*768
    float* ctx      = qbuf + (size_t)BATCH * E_DIM;    // 64*768
    float* attn_out = ctx + (size_t)BATCH * E_DIM;     // 64*768
    (void)ws_size; (void)in_sizes; (void)n_in; (void)out_size;

    // 1. trilinear gather
    sample_kernel<<<dim3(BATCH * NB), dim3(256), 0, stream>>>(x, base, offsets, sampled);

    // 2. sampled @ sample_proj_w.T + b   (2048 x 768 x 768)
    gemm_wmma_f32<<<dim3(E_DIM / 32, MP / 64), dim3(32), 0, stream>>>(
        sampled, sp_w, sp_b, sproj, (int)MP, E_DIM, E_DIM);

    // 3. fused K|V projection: sproj @ [Wk;Wv].T + [bk;bv]  (2048 x 1536 x 768)
    gemm_wmma_f32<<<dim3(2 * E_DIM / 32, MP / 64), dim3(32), 0, stream>>>(
        sproj, in_w + (size_t)E_DIM * E_DIM, in_b + E_DIM, kvbuf,
        (int)MP, 2 * E_DIM, E_DIM);

    // 4. q = bio_embed @ Wq.T + bq  (64 x 768 x 768)
    gemm_wmma_f32<<<dim3(E_DIM / 32, BATCH / 64), dim3(32), 0, stream>>>(
        bio_embed, in_w, in_b, qbuf, BATCH, E_DIM, E_DIM);

    // 5. per-(b,h) softmax attention
    attn_kernel<<<dim3(BATCH * H_HEADS), dim3(32), 0, stream>>>(qbuf, kvbuf, ctx);

    // 6. attn_out = ctx @ out_proj_w.T + b  (64 x 768 x 768)
    gemm_wmma_f32<<<dim3(E_DIM / 32, BATCH / 64), dim3(32), 0, stream>>>(
        ctx, out_w, out_b, attn_out, BATCH, E_DIM, E_DIM);

    // 7. broadcast * confidence -> (64, 513, 768)  [100 MB streaming store]
    const int inner = FULLN * (E_DIM / 4);
    bcast_kernel<<<dim3((inner + 255) / 256, BATCH), dim3(256), 0, stream>>>(
        attn_out, confidence, out);
}